// LightGCNModel_63668595196344
// MI455X (gfx1250) — compile-verified
//
#include <hip/hip_runtime.h>
#include <stdint.h>

// ---------------------------------------------------------------------------
// LightGCN propagation for MI455X (gfx1250).
//
// Bottleneck analysis: 1.84 GFLOP total vs ~2.5 GB of gather/scatter traffic
// per SpMM layer. Dense matrix (150000 x 64 fp32 = 38.4 MB) is L2-resident
// (192 MB L2).  => L2-bandwidth / fp32-atomic bound, WMMA inapplicable
// (random scatter rows).  CDNA5 features used instead:
//   * TDM tensor_load_to_lds: double-buffered DMA of COO edge tiles into LDS,
//     tracked with TENSORcnt (s_wait_tensorcnt), overlapping edge-stream
//     latency with the gather/scatter work.
//   * global_prefetch (__builtin_prefetch) on data-dependent gather rows.
//   * native global_atomic_add_f32 scatter (unsafeAtomicAdd).
//   * wave32 layout: one 64-float embedding row per wave (2 floats/lane),
//     fully coalesced 128B accesses.
// The 1/(L+1)=0.25 mean factor is folded into init/accum (out accumulates
// 0.25*layer_k), eliminating a separate full-output scaling pass.
// ---------------------------------------------------------------------------

#define NUSERS 100000
#define NITEMS 50000
#define NTOT   (NUSERS + NITEMS)
#define DFEAT  64
#define TILE   256   // edges per LDS tile (NNZ=4.8M is divisible by 256)

typedef unsigned int u32x4 __attribute__((ext_vector_type(4)));
typedef int          i32x8 __attribute__((ext_vector_type(8)));
typedef int          i32x4 __attribute__((ext_vector_type(4)));

// Issue one TDM load of a 1-D tile of TILE dwords (global -> LDS).
// tensor_dim0 = remaining elements so OOB tail reads are zero-filled by TDM.
// Descriptor per CDNA5 ISA section 8 (D# groups 0/1; groups 2/3 zero => <=2D).
__device__ __forceinline__ void tdm_load_tile(unsigned lds_off,
                                              const void* gptr,
                                              unsigned tensor_len) {
  unsigned long long g = (unsigned long long)(uintptr_t)gptr;
  u32x4 g0;
  g0[0] = 1u;                                               // count=1 (valid), user mode
  g0[1] = lds_off;                                          // lds_addr (bytes)
  g0[2] = (unsigned)(g & 0xFFFFFFFFull);                    // global_addr[31:0]
  g0[3] = ((unsigned)(g >> 32) & 0x01FFFFFFu) | (2u << 30); // addr[56:32] | type=2
  i32x8 g1;
  g1[0] = (int)(2u << 16);                                  // data_size=4B, no mcast/pad
  g1[1] = (int)((tensor_len & 0xFFFFu) << 16);              // tensor_dim0[15:0]
  g1[2] = (int)(((tensor_len >> 16) & 0xFFFFu) | (1u << 16)); // td0[31:16] | td1=1
  g1[3] = (int)((unsigned)TILE << 16);                      // td1 hi=0 | tile_dim0=TILE
  g1[4] = 1;                                                // tile_dim1=1, tile_dim2=0
  g1[5] = (int)TILE;                                        // td0_stride lo (single row)
  g1[6] = 0;                                                // td0_stride hi | td1_stride lo
  g1[7] = 0;                                                // td1_stride hi
  i32x4 z4 = {0, 0, 0, 0};                                  // groups 2/3 unused (<=2D)
  i32x8 z8 = {0, 0, 0, 0, 0, 0, 0, 0};
  __builtin_amdgcn_tensor_load_to_lds(g0, g1, z4, z4, z8, 0);
}

// out = 0.25*emb0; buf0 = emb0; buf1 = 0 (deterministic ws state every call).
__global__ void __launch_bounds__(256)
lgcn_init_kernel(const float* __restrict__ ue, const float* __restrict__ ie,
                 float* __restrict__ out, float* __restrict__ buf0,
                 float* __restrict__ buf1, unsigned n, unsigned userElems) {
  unsigned i = blockIdx.x * blockDim.x + threadIdx.x;
  if (i >= n) return;
  float v = (i < userElems) ? ue[i] : ie[i - userElems];
  out[i]  = 0.25f * v;
  buf0[i] = v;
  buf1[i] = 0.0f;
}

// y += A_coo @ x  via fp32 HW atomics.  Edge tiles staged into LDS by the TDM,
// double-buffered, TENSORcnt-synchronized.
__global__ void __launch_bounds__(256)
lgcn_spmm_kernel(const int* __restrict__ rows, const int* __restrict__ cols,
                 const float* __restrict__ vals, const float* __restrict__ x,
                 float* __restrict__ y, int nnz, int ntiles) {
  __shared__ int   s_row[2][TILE];
  __shared__ int   s_col[2][TILE];
  __shared__ float s_val[2][TILE];

  const int tpb = (ntiles + (int)gridDim.x - 1) / (int)gridDim.x;
  int t0 = (int)blockIdx.x * tpb;
  int t1 = t0 + tpb;
  if (t1 > ntiles) t1 = ntiles;
  if (t0 >= t1) return;                       // block-uniform early exit

  const int  lane   = (int)(threadIdx.x & 31u);
  const int  wv     = (int)(threadIdx.x >> 5u);
  const bool issuer = (wv == 0);              // wave 0 drives the TDM pipeline

  if (issuer) {                               // prologue: prefetch tile t0 -> buf 0
    const int base = t0 * TILE;
    const unsigned rem = (unsigned)(nnz - base);
    tdm_load_tile((unsigned)(uintptr_t)&s_row[0][0], rows + base, rem);
    tdm_load_tile((unsigned)(uintptr_t)&s_col[0][0], cols + base, rem);
    tdm_load_tile((unsigned)(uintptr_t)&s_val[0][0], vals + base, rem);
  }

  for (int t = t0; t < t1; ++t) {
    const int cur = (t - t0) & 1;
    if (issuer) {
      if (t + 1 < t1) {                       // issue tile t+1 into other buffer
        const int nbase = (t + 1) * TILE;
        const unsigned nrem = (unsigned)(nnz - nbase);
        const int nb = cur ^ 1;
        tdm_load_tile((unsigned)(uintptr_t)&s_row[nb][0], rows + nbase, nrem);
        tdm_load_tile((unsigned)(uintptr_t)&s_col[nb][0], cols + nbase, nrem);
        tdm_load_tile((unsigned)(uintptr_t)&s_val[nb][0], vals + nbase, nrem);
        __builtin_amdgcn_s_wait_tensorcnt(3); // TDM in-order: tile t complete
      } else {
        __builtin_amdgcn_s_wait_tensorcnt(0);
      }
    }
    __syncthreads();                          // LDS tile t visible to all waves

    const int ebase = wv * 32;                // this wave's 32 edges
    {                                         // prefetch gather rows (1 row/lane)
      const int pc = s_col[cur][ebase + lane];
      __builtin_prefetch(x + (unsigned)pc * DFEAT, 0, 3);
    }
#pragma unroll 4
    for (int k = 0; k < 32; ++k) {
      const int e = ebase + k;
      const int   r = s_row[cur][e];
      const int   c = s_col[cur][e];
      const float v = s_val[cur][e];          // zero-filled tail => no-op adds
      const float* xp = x + (unsigned)c * DFEAT;
      float*       yp = y + (unsigned)r * DFEAT;
      const float a = v * xp[lane];           // coalesced 128B per wave
      const float b = v * xp[lane + 32];
      unsafeAtomicAdd(yp + lane,      a);     // global_atomic_add_f32, L2-resident
      unsafeAtomicAdd(yp + lane + 32, b);
    }
    __syncthreads();                          // tile t consumed before reuse
  }
}

// out += 0.25*add; tozero = 0 (prepares next layer's destination buffer).
__global__ void __launch_bounds__(256)
lgcn_accum_zero_kernel(float* __restrict__ out, const float* __restrict__ add,
                       float* __restrict__ tozero, unsigned n) {
  unsigned i = blockIdx.x * blockDim.x + threadIdx.x;
  if (i >= n) return;
  out[i] = fmaf(add[i], 0.25f, out[i]);
  tozero[i] = 0.0f;
}

extern "C" void kernel_launch(void* const* d_in, const int* in_sizes, int n_in,
                              void* d_out, int out_size, void* d_ws, size_t ws_size,
                              hipStream_t stream) {
  const float* user_emb = (const float*)d_in[0];
  const float* item_emb = (const float*)d_in[1];
  const int*   adj_row  = (const int*)d_in[2];
  const int*   adj_col  = (const int*)d_in[3];
  const float* adj_val  = (const float*)d_in[4];
  const int    nnz      = in_sizes[2];

  float* out = (float*)d_out;
  const unsigned nelem     = (unsigned)NTOT * DFEAT;     // 9.6M floats
  const unsigned userElems = (unsigned)NUSERS * DFEAT;

  float* buf0 = (float*)d_ws;                            // 38.4 MB
  float* buf1 = buf0 + (size_t)nelem;                    // 38.4 MB

  const dim3 blk(256);
  const dim3 gElem((nelem + 255u) / 256u);

  lgcn_init_kernel<<<gElem, blk, 0, stream>>>(user_emb, item_emb, out, buf0,
                                              buf1, nelem, userElems);

  const int ntiles = (nnz + TILE - 1) / TILE;            // 18750
  const int gs = ntiles < 2048 ? ntiles : 2048;

  float* src = buf0;
  float* dst = buf1;                                     // zeroed by init
  for (int layer = 0; layer < 3; ++layer) {
    lgcn_spmm_kernel<<<dim3(gs), blk, 0, stream>>>(adj_row, adj_col, adj_val,
                                                   src, dst, nnz, ntiles);
    // out += 0.25*dst; zero src so it can serve as next layer's destination
    lgcn_accum_zero_kernel<<<gElem, blk, 0, stream>>>(out, dst, src, nelem);
    float* tmp = src; src = dst; dst = tmp;
  }
}